// PtychoAD_69896297775162
// MI455X (gfx1250) — compile-verified
//
#include <hip/hip_runtime.h>

#define PI2F 6.28318530717958647692f
#define NN 128
#define NE (NN*NN)
#define PITCH 132            // LDS row pitch: stride mod 64 banks = 4 -> conflict-free b64 ops
#define NEP (NN*PITCH)
#define OM 2
#define PM 4
#define NZS 8
#define NOBJ 512
#define BATCH 64

typedef float v2f __attribute__((ext_vector_type(2)));
typedef float v8f __attribute__((ext_vector_type(8)));

// One DFT "step": D = M * B, M (symmetric DFT matrix) in GLOBAL (row-major,
// stride NN), B/D in LDS with convention logical(r,c) = buf[c*PITCH + r].
// The row-major store of D is then a transposed store, so
//   step(P->W); step(W->P)  computes  P <- M * P * M  with orientation
// restored.  Complex arith: Dr = Mr*Br + (-Mi)*Bi ; Di = Mr*Bi + Mi*Br
// (f32 WMMA NEG[0:1] must be 0 -> VALU-negate the imag A fragment).
// Register blocking: 4 N-tiles per half (8 v8f accs = 64 VGPRs live) to stay
// spill-free; each A fragment pair feeds 16 WMMAs.
__device__ __forceinline__ void cmm(const float* __restrict__ Ar,
                                    const float* __restrict__ Ai,
                                    const float* Br, const float* Bi,
                                    float* Dr, float* Di)
{
    const int tid  = threadIdx.x;
    const int lane = tid & 31;
    const int wv   = tid >> 5;            // 8 waves -> 8 tile rows (M)
    const int m    = lane & 15;           // M (A frag) / N (B frag) in tile
    const int kk   = (lane >> 4) << 1;    // lanes 16-31 hold K+2,K+3
    const int hi8  = (lane >> 4) << 3;    // C/D: lanes 16-31 hold rows M+8..
    const int mrow = wv << 4;

    const float* aR = Ar + (mrow + m)*NN + kk;      // global, +16B per K-step
    const float* aI = Ai + (mrow + m)*NN + kk;
    const float* bR = Br + m*PITCH + kk;            // LDS, K-pair contiguous
    const float* bI = Bi + m*PITCH + kk;

    const v8f vzero = {0.f,0.f,0.f,0.f,0.f,0.f,0.f,0.f};

#pragma unroll
    for (int jh = 0; jh < 2; ++jh) {                // two halves of 4 N-tiles
        v8f accR[4], accI[4];
#pragma unroll
        for (int j = 0; j < 4; ++j) { accR[j] = vzero; accI[j] = vzero; }

        const float* bRh = bR + jh*64*PITCH;
        const float* bIh = bI + jh*64*PITCH;

#pragma clang loop unroll_count(2)
        for (int k0 = 0; k0 < NN; k0 += 4) {
            v2f ar = *(const v2f*)(aR + k0);
            v2f ai = *(const v2f*)(aI + k0);
            v2f am; am.x = -ai.x; am.y = -ai.y;
#pragma unroll
            for (int j = 0; j < 4; ++j) {
                v2f br = *(const v2f*)(bRh + j*16*PITCH + k0);
                v2f bi = *(const v2f*)(bIh + j*16*PITCH + k0);
                accR[j] = __builtin_amdgcn_wmma_f32_16x16x4_f32(false, ar, false, br, (short)0, accR[j], false, false);
                accR[j] = __builtin_amdgcn_wmma_f32_16x16x4_f32(false, am, false, bi, (short)0, accR[j], false, false);
                accI[j] = __builtin_amdgcn_wmma_f32_16x16x4_f32(false, ar, false, bi, (short)0, accI[j], false, false);
                accI[j] = __builtin_amdgcn_wmma_f32_16x16x4_f32(false, ai, false, br, (short)0, accI[j], false, false);
            }
        }
#pragma unroll
        for (int j = 0; j < 4; ++j) {
            const int col = (jh*4 + j)*16 + m;
#pragma unroll
            for (int r = 0; r < 8; ++r) {
                Dr[(mrow + hi8 + r)*PITCH + col] = accR[j][r];   // transposed store
                Di[(mrow + hi8 + r)*PITCH + col] = accI[j][r];
            }
        }
    }
}

__global__ void k_zero(float* p, int n) {
    int i = blockIdx.x * blockDim.x + threadIdx.x;
    if (i < n) p[i] = 0.0f;
}

// DFT matrices: F = exp(-2πi·uy/128); G = conj(F)/128 (ifft norm folded in).
__global__ void k_dft(float* Fr, float* Fi, float* Gr, float* Gi) {
    int i = blockIdx.x * 256 + threadIdx.x;        // 16384 elements
    int u = i >> 7, y = i & 127;
    float ang = (PI2F / 128.0f) * (float)((u * y) & 127);
    float c = cosf(ang), s = sinf(ang);
    Fr[i] = c;  Fi[i] = -s;
    Gr[i] = c * (1.0f/128.0f);  Gi[i] = s * (1.0f/128.0f);
}

// Phat[p] = F * probe[p] * F  (one workgroup per probe mode)
__global__ void __launch_bounds__(256, 1)
k_probehat(const float* __restrict__ probe,
           const float* __restrict__ Fr, const float* __restrict__ Fi,
           float* __restrict__ PhatR, float* __restrict__ PhatI)
{
    extern __shared__ float sm[];
    float* pR = sm;         float* pI = sm + NEP;
    float* wR = sm + 2*NEP; float* wI = sm + 3*NEP;
    const int p = blockIdx.x;
    for (int i = threadIdx.x; i < NE; i += blockDim.x) {
        int y = i >> 7, x = i & 127;
        pR[x*PITCH + y] = probe[(p*NE + i)*2 + 0];
        pI[x*PITCH + y] = probe[(p*NE + i)*2 + 1];
    }
    __syncthreads();
    cmm(Fr, Fi, pR, pI, wR, wI);  __syncthreads();
    cmm(Fr, Fi, wR, wI, pR, pI);  __syncthreads();
    for (int i = threadIdx.x; i < NE; i += blockDim.x) {
        int y = i >> 7, x = i & 127;
        PhatR[p*NE + i] = pR[x*PITCH + y];
        PhatI[p*NE + i] = pI[x*PITCH + y];
    }
}

// One workgroup per (n,o,p): shifted probe -> 8-slice multislice -> |FFT2|^2.
__global__ void __launch_bounds__(256, 1)
k_chain(const float* __restrict__ obja, const float* __restrict__ objp,
        const float* __restrict__ pshift, const float* __restrict__ H,
        const float* __restrict__ occu,
        const int* __restrict__ indices, const int* __restrict__ crop,
        const float* __restrict__ Fr, const float* __restrict__ Fi,
        const float* __restrict__ Gr, const float* __restrict__ Gi,
        const float* __restrict__ PhatR, const float* __restrict__ PhatI,
        float* __restrict__ dp)
{
    extern __shared__ float sm[];
    float* pR = sm;         float* pI = sm + NEP;
    float* wR = sm + 2*NEP; float* wI = sm + 3*NEP;

    const int b = blockIdx.x;
    const int p = b & 3;
    const int o = (b >> 2) & 1;
    const int n = b >> 3;
    const int idx = indices[n];
    const int cy = crop[idx*2 + 0], cx = crop[idx*2 + 1];
    const float s0 = pshift[idx*2 + 0], s1 = pshift[idx*2 + 1];
    const float w_o = occu[o];

    // psi = ifft2(Phat[p] .* ramp(n))  (sub-pixel shifted probe)
    for (int i = threadIdx.x; i < NE; i += blockDim.x) {
        int y = i >> 7, x = i & 127;
        float ky = (float)((y < 64) ? y : y - 128) * (1.0f/128.0f);
        float kx = (float)((x < 64) ? x : x - 128) * (1.0f/128.0f);
        float ph = -PI2F * (ky * s0 + kx * s1);
        float c = cosf(ph), s = sinf(ph);
        float re = PhatR[p*NE + i], im = PhatI[p*NE + i];
        pR[x*PITCH + y] = re*c - im*s;
        pI[x*PITCH + y] = re*s + im*c;
    }
    __syncthreads();
    cmm(Gr, Gi, pR, pI, wR, wI);  __syncthreads();
    cmm(Gr, Gi, wR, wI, pR, pI);  __syncthreads();

    for (int z = 0; z < NZS; ++z) {
        // psi *= T(n,o,z) = a * exp(i*phase), cropped from 512x512 object
        for (int i = threadIdx.x; i < NE; i += blockDim.x) {
            int y = i >> 7, x = i & 127;
            long oidx = ((long)(o*NZS + z)*NOBJ + (cy + y))*NOBJ + (cx + x);
            float a = obja[oidx], ph = objp[oidx];
            float tr = a * cosf(ph), ti = a * sinf(ph);
            int li = x*PITCH + y;
            float re = pR[li], im = pI[li];
            pR[li] = re*tr - im*ti;
            pI[li] = re*ti + im*tr;
        }
        __syncthreads();
        if (z < NZS - 1) {
            // psi = ifft2(fft2(psi) .* H)
            cmm(Fr, Fi, pR, pI, wR, wI);  __syncthreads();
            cmm(Fr, Fi, wR, wI, pR, pI);  __syncthreads();
            for (int i = threadIdx.x; i < NE; i += blockDim.x) {
                int y = i >> 7, x = i & 127, li = x*PITCH + y;
                float hr = H[i*2 + 0], hi = H[i*2 + 1];
                float re = pR[li], im = pI[li];
                pR[li] = re*hr - im*hi;
                pI[li] = re*hi + im*hr;
            }
            __syncthreads();
            cmm(Gr, Gi, pR, pI, wR, wI);  __syncthreads();
            cmm(Gr, Gi, wR, wI, pR, pI);  __syncthreads();
        }
    }

    // dp[n] += occu[o] * |fft2(psi)|^2
    cmm(Fr, Fi, pR, pI, wR, wI);  __syncthreads();
    cmm(Fr, Fi, wR, wI, pR, pI);  __syncthreads();
    for (int i = threadIdx.x; i < NE; i += blockDim.x) {
        int y = i >> 7, x = i & 127, li = x*PITCH + y;
        float re = pR[li], im = pI[li];
        atomicAdd(&dp[n*NE + i], w_o * (re*re + im*im));
    }
}

extern "C" void kernel_launch(void* const* d_in, const int* in_sizes, int n_in,
                              void* d_out, int out_size, void* d_ws, size_t ws_size,
                              hipStream_t stream) {
    const float* obja   = (const float*)d_in[0];
    const float* objp   = (const float*)d_in[1];
    const float* probe  = (const float*)d_in[2];
    const float* pshift = (const float*)d_in[3];
    const float* H      = (const float*)d_in[4];
    const float* occu   = (const float*)d_in[5];
    const int*   indices= (const int*)d_in[6];
    const int*   crop   = (const int*)d_in[7];
    float* dp = (float*)d_out;

    float* ws = (float*)d_ws;
    float* Fr = ws;            float* Fi = ws + NE;
    float* Gr = ws + 2*NE;     float* Gi = ws + 3*NE;
    float* PhatR = ws + 4*NE;  float* PhatI = ws + 4*NE + PM*NE;   // ~786 KB total

    const size_t shbytes = (size_t)4 * NEP * sizeof(float);        // ~264 KB LDS
    hipFuncSetAttribute(reinterpret_cast<const void*>(k_probehat),
                        hipFuncAttributeMaxDynamicSharedMemorySize, (int)shbytes);
    hipFuncSetAttribute(reinterpret_cast<const void*>(k_chain),
                        hipFuncAttributeMaxDynamicSharedMemorySize, (int)shbytes);

    k_zero<<<(out_size + 255)/256, 256, 0, stream>>>(dp, out_size);
    k_dft<<<NE/256, 256, 0, stream>>>(Fr, Fi, Gr, Gi);
    k_probehat<<<PM, 256, shbytes, stream>>>(probe, Fr, Fi, PhatR, PhatI);
    k_chain<<<BATCH*OM*PM, 256, shbytes, stream>>>(obja, objp, pshift, H, occu,
                                                   indices, crop, Fr, Fi, Gr, Gi,
                                                   PhatR, PhatI, dp);
}